// LLamaAttentionBlock_30167850287808
// MI455X (gfx1250) — compile-verified
//
#include <hip/hip_runtime.h>
#include <hip/hip_bf16.h>
#include <math.h>

// Problem constants (match reference)
#define NB 16       // batches
#define NH 16       // heads
#define ND 128      // head dim
#define BLKT 16     // tokens per cache block
#define MAXB 128    // blocks per sequence
#define MAXS 2048   // max sequence length
#define NSPLIT 16   // sequence splits (flash-decode)
#define CBLKS (MAXB / NSPLIT)     // 8 cache blocks per chunk
#define CTOK  (CBLKS * BLKT)      // 128 tokens per chunk
#define PSTRIDE 130               // partial: 128 acc + m + l
#define SOFTMAX_SCALE 0.08838834764831845f  // 1/sqrt(128)

typedef float v2f __attribute__((ext_vector_type(2)));
typedef float v4f __attribute__((ext_vector_type(4)));
typedef float v8f __attribute__((ext_vector_type(8)));

// D(16x16,f32) = A(16x4,f32) x B(4x16,f32) + C  -- CDNA5 wave32 WMMA
__device__ inline v8f wmma_f32_16x16x4(v2f a, v2f b, v8f c) {
  return __builtin_amdgcn_wmma_f32_16x16x4_f32(
      /*neg_a=*/false, a, /*neg_b=*/false, b,
      /*c_mod=*/(short)0, c, /*reuse_a=*/false, /*reuse_b=*/false);
}

// d-dimension assigned to (wmma step t, k-slot half*2+e) so that each lane's
// B-fragment source is one contiguous float4: lane-half hh, step pair t2:
//   d(t=2*t2+p, e, hh) = 8*t2 + 4*hh + 2*p + e
__device__ inline int dmap(int t, int e, int half) {
  return 8 * (t >> 1) + 4 * half + 2 * (t & 1) + e;
}

// One wave per (batch, head); one workgroup per (batch, chunk of 128 tokens).
// Flash-attention partials (m, l, acc[128]) written to workspace.
__global__ __launch_bounds__(512)
void attn_partial_kernel(const float* __restrict__ Q,
                         const float* __restrict__ Kf,
                         const float* __restrict__ Vf,
                         const float* __restrict__ Kcache,
                         const float* __restrict__ Vcache,
                         const float* __restrict__ cosp,
                         const float* __restrict__ sinp,
                         const float* __restrict__ maskp,
                         const int*  __restrict__ input_length,
                         const int*  __restrict__ block_tables,
                         float* __restrict__ ws)
{
  const int lane = threadIdx.x & 31;
  const int h    = threadIdx.x >> 5;   // wave id == head
  const int half = lane >> 4;          // 0: lanes 0-15, 1: lanes 16-31
  const int col  = lane & 15;          // token column / d column within tile
  const int b    = blockIdx.x / NSPLIT;
  const int c    = blockIdx.x % NSPLIT;
  const int len  = input_length[b];
  const int cstart = c * CTOK;

  float* part = ws + ((size_t)(b * NH + h) * NSPLIT + c) * PSTRIDE;

  if (cstart >= len) {  // empty chunk: write neutral partial
    if (half == 0) {
      #pragma unroll
      for (int dt = 0; dt < 8; ++dt) part[dt * 16 + col] = 0.0f;
    }
    if (lane == 0) { part[128] = -3.0e38f; part[129] = 0.0f; }
    return;
  }

  const float* cb = cosp + (size_t)b * ND;
  const float* sb = sinp + (size_t)b * ND;
  const float* qb = Q  + (size_t)(b * NH + h) * ND;
  const float* kb = Kf + (size_t)(b * NH + h) * ND;
  const float* vb = Vf + (size_t)(b * NH + h) * ND;
  const int* btab = block_tables + b * MAXB;

  // Prologue: RoPE'd Q fragments (permuted d-mapping) and the fresh-token
  // score q_rope . k_rope (computed once; patched into its column later).
  v2f aq[32];
  float sf_part = 0.0f;
  #pragma unroll
  for (int t = 0; t < 32; ++t) {
    #pragma unroll
    for (int e = 0; e < 2; ++e) {
      const int d = dmap(t, e, half);
      const float qx   = qb[d];
      const float qrot = (d < 64) ? -qb[d + 64] : qb[d - 64];
      const float qv   = qx * cb[d] + qrot * sb[d];
      aq[t][e] = qv;
      const float kx   = kb[d];
      const float krot = (d < 64) ? -kb[d + 64] : kb[d - 64];
      sf_part += qv * (kx * cb[d] + krot * sb[d]);
    }
  }
  // each half holds the partial over its 64 dims; combine across halves
  const float sfresh = sf_part + __shfl_xor(sf_part, 16, 32);

  float m_run = -3.0e38f;
  float l_run = 0.0f;
  float accd[8];
  #pragma unroll
  for (int dt = 0; dt < 8; ++dt) accd[dt] = 0.0f;

  for (int blk = 0; blk < CBLKS; ++blk) {
    const int tb = cstart + blk * BLKT;
    if (tb >= len) break;                       // wave-uniform
    const int bt = btab[tb >> 4];
    const int sg = tb + col;                    // this lane's token column

    const float* krow =
        Kcache + ((size_t)(bt * BLKT + col)) * (NH * ND) + (size_t)h * ND;

    // Prefetch next block's K/V rows (global_prefetch_b8)
    if (blk + 1 < CBLKS && tb + BLKT < len) {
      const int btn = btab[(tb >> 4) + 1];
      const size_t roff = ((size_t)(btn * BLKT + col)) * (NH * ND) + (size_t)h * ND;
      __builtin_prefetch(Kcache + roff, 0, 1);
      __builtin_prefetch(Vcache + roff, 0, 1);
    }

    // ---- QK^T phase 1: batched B128 loads into a register tile ----
    v4f kb4[16];
    #pragma unroll
    for (int t2 = 0; t2 < 16; ++t2)
      kb4[t2] = *(const v4f*)(krow + 8 * t2 + 4 * half);

    // ---- QK^T phase 2: 32 chained f32 WMMAs (K=4 dims each) ----
    v8f c8 = {0.f, 0.f, 0.f, 0.f, 0.f, 0.f, 0.f, 0.f};
    #pragma unroll
    for (int t2 = 0; t2 < 16; ++t2) {
      v2f b0; b0[0] = kb4[t2][0]; b0[1] = kb4[t2][1];
      v2f b1; b1[0] = kb4[t2][2]; b1[1] = kb4[t2][3];
      c8 = wmma_f32_16x16x4(aq[2 * t2],     b0, c8);
      c8 = wmma_f32_16x16x4(aq[2 * t2 + 1], b1, c8);
    }

    // C rows are replicated (A was row-broadcast): score[col] = any row.
    float raw = (sg == len - 1) ? sfresh : c8[0];   // patch fresh column
    float s = raw * SOFTMAX_SCALE + maskp[(size_t)b * MAXS + sg];
    if (sg >= len) s = -1.0e30f;

    // Online softmax: max/sum over the 16 columns (halves are duplicates).
    float tmax = s;
    #pragma unroll
    for (int off = 8; off >= 1; off >>= 1)
      tmax = fmaxf(tmax, __shfl_xor(tmax, off, 32));
    const float m_new = fmaxf(m_run, tmax);
    const float corr  = __expf(m_run - m_new);
    const float p     = __expf(s - m_new);
    float tsum = p;
    #pragma unroll
    for (int off = 8; off >= 1; off >>= 1)
      tsum += __shfl_xor(tsum, off, 32);
    l_run = l_run * corr + tsum;
    m_run = m_new;

    // A2 fragments for PV: rows broadcast p, K indexes tokens (4 per WMMA).
    v2f a2[4];
    #pragma unroll
    for (int j = 0; j < 4; ++j) {
      a2[j][0] = __shfl(p, 4 * j + half * 2,     32);
      a2[j][1] = __shfl(p, 4 * j + half * 2 + 1, 32);
    }

    // ---- PV: out[d] += sum_s p[s] * V[s][d], 8 d-tiles x 4 token-steps ----
    // V rows read straight from cache; the (stale) row at the fresh slot is
    // cancelled exactly by the post-loop correction below.
    const float* vrow0 =
        Vcache + ((size_t)bt * BLKT + half * 2) * (NH * ND) + (size_t)h * ND;
    #pragma unroll
    for (int dt = 0; dt < 8; ++dt) {
      v8f cp = {0.f, 0.f, 0.f, 0.f, 0.f, 0.f, 0.f, 0.f};
      #pragma unroll
      for (int j = 0; j < 4; ++j) {
        const float* vr = vrow0 + (size_t)(4 * j) * (NH * ND);
        v2f b2;
        b2[0] = vr[dt * 16 + col];
        b2[1] = vr[(size_t)(NH * ND) + dt * 16 + col];
        cp = wmma_f32_16x16x4(a2[j], b2, cp);
      }
      accd[dt] = accd[dt] * corr + cp[0];
    }
  }

  // Fresh-token V correction: if this chunk contains pos = len-1, the PV loop
  // used the stale cache row there; swap its contribution for the fresh V.
  const int pos = len - 1;
  if (pos >= cstart && pos < cstart + CTOK) {
    const int btp = btab[pos >> 4];
    const float* vstale =
        Vcache + ((size_t)(btp * BLKT + (pos & 15))) * (NH * ND) + (size_t)h * ND;
    const float s_pos = sfresh * SOFTMAX_SCALE + maskp[(size_t)b * MAXS + pos];
    const float p_pos = __expf(s_pos - m_run);
    #pragma unroll
    for (int dt = 0; dt < 8; ++dt) {
      const int d = dt * 16 + col;
      accd[dt] += p_pos * (vb[d] - vstale[d]);
    }
  }

  if (half == 0) {
    #pragma unroll
    for (int dt = 0; dt < 8; ++dt) part[dt * 16 + col] = accd[dt];
  }
  if (lane == 0) { part[128] = m_run; part[129] = l_run; }
}

// One wave per (b,h): log-sum-exp merge of the NSPLIT partials.
__global__ __launch_bounds__(256)
void attn_combine_kernel(const float* __restrict__ ws, float* __restrict__ out)
{
  const int lane = threadIdx.x & 31;
  const int pair = blockIdx.x * 8 + (threadIdx.x >> 5);
  if (pair >= NB * NH) return;
  const float* base = ws + (size_t)pair * NSPLIT * PSTRIDE;

  float M = -3.0e38f;
  #pragma unroll
  for (int cc = 0; cc < NSPLIT; ++cc)
    M = fmaxf(M, base[cc * PSTRIDE + 128]);

  float L = 0.0f;
  float o0 = 0.f, o1 = 0.f, o2 = 0.f, o3 = 0.f;
  #pragma unroll
  for (int cc = 0; cc < NSPLIT; ++cc) {
    const float* pc = base + cc * PSTRIDE;
    const float w = __expf(pc[128] - M);
    L  += w * pc[129];
    o0 += w * pc[lane];
    o1 += w * pc[lane + 32];
    o2 += w * pc[lane + 64];
    o3 += w * pc[lane + 96];
  }
  const float inv = 1.0f / L;  // len >= 1 guarantees L > 0
  float* op = out + (size_t)pair * ND;
  op[lane]      = o0 * inv;
  op[lane + 32] = o1 * inv;
  op[lane + 64] = o2 * inv;
  op[lane + 96] = o3 * inv;
}

extern "C" void kernel_launch(void* const* d_in, const int* in_sizes, int n_in,
                              void* d_out, int out_size, void* d_ws, size_t ws_size,
                              hipStream_t stream) {
  const float* Q       = (const float*)d_in[0];
  const float* K       = (const float*)d_in[1];
  const float* V       = (const float*)d_in[2];
  const float* Kcache  = (const float*)d_in[3];
  const float* Vcache  = (const float*)d_in[4];
  const float* cosp    = (const float*)d_in[5];
  const float* sinp    = (const float*)d_in[6];
  const float* maskp   = (const float*)d_in[7];
  const int*   inlen   = (const int*)d_in[8];
  // d_in[9]  = save_slots (int64): not needed -- the fresh token is handled
  //            in-kernel at position len-1 (we must not mutate input caches).
  const int*   btables = (const int*)d_in[10];
  // d_in[11] = max_s scalar: compile-time constant here.
  float* ws   = (float*)d_ws;   // needs NB*NH*NSPLIT*130*4 = ~2.03 MB
  float* outf = (float*)d_out;

  attn_partial_kernel<<<dim3(NB * NSPLIT), dim3(512), 0, stream>>>(
      Q, K, V, Kcache, Vcache, cosp, sinp, maskp, inlen, btables, ws);
  attn_combine_kernel<<<dim3((NB * NH) / 8), dim3(256), 0, stream>>>(ws, outf);
}